// SimpleRobotGraph_16484084482983
// MI455X (gfx1250) — compile-verified
//
#include <hip/hip_runtime.h>
#include <hip/hip_bf16.h>

typedef float v2f __attribute__((ext_vector_type(2)));
typedef float v4f __attribute__((ext_vector_type(4)));
typedef float v8f __attribute__((ext_vector_type(8)));

#define NJ 7
#define ND 64
#define CSTR 132   // combined row stride (floats): 128 + 4 pad -> conflict-free ds_load_b64
#define W1STR 132  // W1 row stride (floats): 128 + 4 pad
#define W2STR 68   // W2 row stride (floats): 64 + 4 pad
#define HSTR 68    // h row stride (floats): 64 + 4 pad

// D = A(16x4 f32) * B(4x16 f32) + C(16x16 f32), fp32 matrix pipe.
// A frag: lanes 0-15 rows M=0..15 with K={0,1}; lanes 16-31 same rows, K={2,3}.
// B frag (symmetric): lanes 0-15 cols N=0..15 with K={0,1}; lanes 16-31 K={2,3}.
// C/D: VGPR v, lanes 0-15 -> row v, lanes 16-31 -> row v+8; col = lane&15.

__global__ __launch_bounds__(128)
void robot_graph_mlp_wmma(const float* __restrict__ x,   // [nrows][64]
                          const float* __restrict__ W1,  // [64][128] (out,in)
                          const float* __restrict__ b1,  // [64]
                          const float* __restrict__ W2,  // [64][64]
                          const float* __restrict__ b2,  // [64]
                          float* __restrict__ out,       // [nrows][64]
                          int nrows) {
  __shared__ float sComb[64 * CSTR];      // combined activations, 64 rows
  __shared__ float sW1[64 * W1STR];
  __shared__ float sW2[64 * W2STR];
  __shared__ float sH[4][16 * HSTR];      // per-wave hidden tile (transpose staging)
  __shared__ float sB1[64];
  __shared__ float sB2[64];

  const int tid = threadIdx.x;
  const int R0  = blockIdx.x * 64;

  // ---- stage weights & biases (coalesced b128) ----
  for (int i = tid; i < 64 * 32; i += 128) {          // W1: 64 rows x 32 float4
    int r = i >> 5, c4 = i & 31;
    v4f v = *(const v4f*)(W1 + r * 128 + c4 * 4);
    *(v4f*)(sW1 + r * W1STR + c4 * 4) = v;
  }
  for (int i = tid; i < 64 * 16; i += 128) {          // W2: 64 rows x 16 float4
    int r = i >> 4, c4 = i & 15;
    v4f v = *(const v4f*)(W2 + r * 64 + c4 * 4);
    *(v4f*)(sW2 + r * W2STR + c4 * 4) = v;
  }
  if (tid < 64) { sB1[tid] = b1[tid]; sB2[tid] = b2[tid]; }

  // ---- stage combined = [x | x + prev + next] for rows R0..R0+63 ----
  for (int i = tid; i < 64 * 16; i += 128) {          // 16 float4 per row
    int r = i >> 4, c4 = i & 15;
    int row = R0 + r;
    int rowc = row < nrows ? row : nrows - 1;
    int j = rowc % NJ;
    v4f c = *(const v4f*)(x + (size_t)rowc * ND + c4 * 4);
    v4f nb = c;
    if (j > 0) {
      v4f p = *(const v4f*)(x + (size_t)(rowc - 1) * ND + c4 * 4);
      nb += p;
    }
    if (j < NJ - 1) {
      v4f n = *(const v4f*)(x + (size_t)(rowc + 1) * ND + c4 * 4);
      nb += n;
    }
    *(v4f*)(sComb + r * CSTR + c4 * 4)      = c;
    *(v4f*)(sComb + r * CSTR + 64 + c4 * 4) = nb;
  }
  __syncthreads();

  // ---- per-wave 16-row tile ----
  const int wave    = tid >> 5;
  const int lane    = tid & 31;
  const int halfsel = lane >> 4;   // 0: K pair {0,1} / rows 0-7 of D; 1: {2,3} / rows 8-15
  const int ln      = lane & 15;
  const int kk      = halfsel * 2;

  // ---- layer 1: [16x128] @ W1^T -> 4 N-tiles, 32 K-steps of 4 ----
  const float* aRow = sComb + (wave * 16 + ln) * CSTR + kk;
  const float* w1L  = sW1 + ln * W1STR + kk;
  v8f a0 = {}, a1 = {}, a2 = {}, a3 = {};
#pragma unroll
  for (int s = 0; s < 32; ++s) {
    v2f a   = *(const v2f*)(aRow + 4 * s);
    v2f bb0 = *(const v2f*)(w1L + (0 * 16) * W1STR + 4 * s);
    v2f bb1 = *(const v2f*)(w1L + (1 * 16) * W1STR + 4 * s);
    v2f bb2 = *(const v2f*)(w1L + (2 * 16) * W1STR + 4 * s);
    v2f bb3 = *(const v2f*)(w1L + (3 * 16) * W1STR + 4 * s);
    a0 = __builtin_amdgcn_wmma_f32_16x16x4_f32(false, a, false, bb0, (short)0, a0, false, false);
    a1 = __builtin_amdgcn_wmma_f32_16x16x4_f32(false, a, false, bb1, (short)0, a1, false, false);
    a2 = __builtin_amdgcn_wmma_f32_16x16x4_f32(false, a, false, bb2, (short)0, a2, false, false);
    a3 = __builtin_amdgcn_wmma_f32_16x16x4_f32(false, a, false, bb3, (short)0, a3, false, false);
  }

  // bias + ReLU in registers, then transpose h through per-wave LDS tile
  {
    float bb0 = sB1[ln], bb1 = sB1[16 + ln], bb2 = sB1[32 + ln], bb3 = sB1[48 + ln];
    float* hW = sH[wave];
#pragma unroll
    for (int v = 0; v < 8; ++v) {
      int m = v + 8 * halfsel;
      hW[m * HSTR + ln +  0] = fmaxf(a0[v] + bb0, 0.0f);
      hW[m * HSTR + ln + 16] = fmaxf(a1[v] + bb1, 0.0f);
      hW[m * HSTR + ln + 32] = fmaxf(a2[v] + bb2, 0.0f);
      hW[m * HSTR + ln + 48] = fmaxf(a3[v] + bb3, 0.0f);
    }
  }
  __syncthreads();

  // ---- layer 2: [16x64] @ W2^T -> 4 N-tiles, 16 K-steps of 4 ----
  const float* a2Row = sH[wave] + ln * HSTR + kk;
  const float* w2L   = sW2 + ln * W2STR + kk;
  v8f o0 = {}, o1 = {}, o2 = {}, o3 = {};
#pragma unroll
  for (int s = 0; s < 16; ++s) {
    v2f a   = *(const v2f*)(a2Row + 4 * s);
    v2f bb0 = *(const v2f*)(w2L + (0 * 16) * W2STR + 4 * s);
    v2f bb1 = *(const v2f*)(w2L + (1 * 16) * W2STR + 4 * s);
    v2f bb2 = *(const v2f*)(w2L + (2 * 16) * W2STR + 4 * s);
    v2f bb3 = *(const v2f*)(w2L + (3 * 16) * W2STR + 4 * s);
    o0 = __builtin_amdgcn_wmma_f32_16x16x4_f32(false, a, false, bb0, (short)0, o0, false, false);
    o1 = __builtin_amdgcn_wmma_f32_16x16x4_f32(false, a, false, bb1, (short)0, o1, false, false);
    o2 = __builtin_amdgcn_wmma_f32_16x16x4_f32(false, a, false, bb2, (short)0, o2, false, false);
    o3 = __builtin_amdgcn_wmma_f32_16x16x4_f32(false, a, false, bb3, (short)0, o3, false, false);
  }

  // ---- bias + store (lanes 0-15 and 16-31 each write contiguous 64B row segments) ----
  {
    float bb0 = sB2[ln], bb1 = sB2[16 + ln], bb2 = sB2[32 + ln], bb3 = sB2[48 + ln];
    int baseRow = R0 + wave * 16 + 8 * halfsel;
#pragma unroll
    for (int v = 0; v < 8; ++v) {
      int row = baseRow + v;
      if (row < nrows) {
        float* po = out + (size_t)row * ND + ln;
        po[0]  = o0[v] + bb0;
        po[16] = o1[v] + bb1;
        po[32] = o2[v] + bb2;
        po[48] = o3[v] + bb3;
      }
    }
  }
}

extern "C" void kernel_launch(void* const* d_in, const int* in_sizes, int n_in,
                              void* d_out, int out_size, void* d_ws, size_t ws_size,
                              hipStream_t stream) {
  const float* x  = (const float*)d_in[0];  // [B,J,64] fp32
  const float* W1 = (const float*)d_in[1];  // [64,128]
  const float* b1 = (const float*)d_in[2];  // [64]
  const float* W2 = (const float*)d_in[3];  // [64,64]
  const float* b2 = (const float*)d_in[4];  // [64]
  float* out = (float*)d_out;

  int nrows = in_sizes[0] / ND;             // B*J = 917504
  int blocks = (nrows + 63) / 64;           // 64 rows per block (4 waves x 16)
  robot_graph_mlp_wmma<<<blocks, 128, 0, stream>>>(x, W1, b1, W2, b2, out, nrows);
}